// MMIN_DGODE_78039555768358
// MI455X (gfx1250) — compile-verified
//
#include <hip/hip_runtime.h>
#include <math.h>

#define B_DIM 2048
#define H_DIM 1024
#define DIN   1920
#define NSTEP 32

typedef __attribute__((ext_vector_type(2))) float v2f;
typedef __attribute__((ext_vector_type(8))) float v8f;

// ---------------------------------------------------------------------------
// Tiled f32 WMMA GEMM:  C[M,N] = op(A[M,K] @ B[K,N] (+C) (+bias)), op=tanh?
// Block tile 128(M) x 64(N), 256 threads = 8 wave32, wave tile 32x32 via
// 2x2 V_WMMA_F32_16X16X4_F32.  K staged through double-buffered LDS in
// chunks of 16, stored K-PAIRED ( [kp][x][2] ) so every WMMA fragment is a
// single aligned ds_load_b64.  Plane strides are ≡32 (mod 64) dwords so the
// upper half-wave (kp+1) hits the opposite 32 LDS banks from the lower half.
// ---------------------------------------------------------------------------
template<bool BIAS, bool ACCUM, bool TANH>
__global__ __launch_bounds__(256)
void gemm_f32_wmma(const float* __restrict__ A, int lda,
                   const float* __restrict__ Bm, int ldb,
                   const float* __restrict__ bias,
                   float* __restrict__ C, int ldc,
                   int M, int N, int K)
{
    // [buf][kp][x][2] ; A plane = 144 pairs (288 dw ≡ 32 mod 64)
    //                   B plane =  80 pairs (160 dw ≡ 32 mod 64)
    __shared__ float As[2][8][144][2];
    __shared__ float Bs[2][8][80][2];

    const int tid  = threadIdx.x;
    const int lane = tid & 31;
    const int wave = tid >> 5;
    const int wm   = wave >> 1;                 // 0..3 -> 32-row strip
    const int wn   = wave & 1;                  // 0..1 -> 32-col strip
    const int mBlock = blockIdx.y * 128;
    const int nBlock = blockIdx.x * 64;

    const int ml = lane & 15;
    const int hi = lane >> 4;                   // upper half-wave: K+2,K+3

    // staging assignments
    const int mA  = tid & 127;                  // A: one thread per tile row
    const int kpA = (tid >> 7) * 4;             //    planes 0-3 or 4-7
    const int kpB = tid >> 5;                   // B: one kp-plane per wave
    const int n2B = (tid & 31) * 2;             //    2 cols per thread

    v8f acc[2][2];
#pragma unroll
    for (int i = 0; i < 2; ++i)
#pragma unroll
        for (int j = 0; j < 2; ++j)
            acc[i][j] = (v8f)(0.0f);

    float4 ra0, ra1;                            // in-flight A tile
    v2f    rb0, rb1;                            // in-flight B tile

    const int nTiles = K >> 4;

    // ---- prologue: fetch tile 0 ----
    {
        const float* pa = A + (size_t)(mBlock + mA) * lda + (kpA * 2);
        ra0 = *(const float4*)(pa);
        ra1 = *(const float4*)(pa + 4);
        const float* pb = Bm + (size_t)(2 * kpB) * ldb + nBlock + n2B;
        rb0 = *(const v2f*)(pb);
        rb1 = *(const v2f*)(pb + ldb);
    }
    {
        *(v2f*)&As[0][kpA + 0][mA][0] = (v2f){ra0.x, ra0.y};
        *(v2f*)&As[0][kpA + 1][mA][0] = (v2f){ra0.z, ra0.w};
        *(v2f*)&As[0][kpA + 2][mA][0] = (v2f){ra1.x, ra1.y};
        *(v2f*)&As[0][kpA + 3][mA][0] = (v2f){ra1.z, ra1.w};
        float4 bq = {rb0.x, rb1.x, rb0.y, rb1.y};
        *(float4*)&Bs[0][kpB][n2B][0] = bq;
    }
    __syncthreads();

    for (int t = 0; t < nTiles; ++t) {
        const int buf  = t & 1;
        const bool more = (t + 1) < nTiles;

        // issue next tile's global loads; latency hidden under the WMMAs
        if (more) {
            const int k0 = (t + 1) * 16;
            const float* pa = A + (size_t)(mBlock + mA) * lda + (k0 + kpA * 2);
            ra0 = *(const float4*)(pa);
            ra1 = *(const float4*)(pa + 4);
            const float* pb = Bm + (size_t)(k0 + 2 * kpB) * ldb + nBlock + n2B;
            rb0 = *(const v2f*)(pb);
            rb1 = *(const v2f*)(pb + ldb);
        }

        // ---- compute on buf ----
#pragma unroll
        for (int kk = 0; kk < 16; kk += 4) {
            const int p = (kk >> 1) + hi;       // pair-plane for this half-wave
            v2f a0 = *(const v2f*)&As[buf][p][wm * 32 + ml][0];
            v2f a1 = *(const v2f*)&As[buf][p][wm * 32 + 16 + ml][0];
            v2f b0 = *(const v2f*)&Bs[buf][p][wn * 32 + ml][0];
            v2f b1 = *(const v2f*)&Bs[buf][p][wn * 32 + 16 + ml][0];
            acc[0][0] = __builtin_amdgcn_wmma_f32_16x16x4_f32(
                false, a0, false, b0, (short)0, acc[0][0], false, false);
            acc[0][1] = __builtin_amdgcn_wmma_f32_16x16x4_f32(
                false, a0, false, b1, (short)0, acc[0][1], false, false);
            acc[1][0] = __builtin_amdgcn_wmma_f32_16x16x4_f32(
                false, a1, false, b0, (short)0, acc[1][0], false, false);
            acc[1][1] = __builtin_amdgcn_wmma_f32_16x16x4_f32(
                false, a1, false, b1, (short)0, acc[1][1], false, false);
        }

        // ---- stage next tile into the other buffer; 1 barrier per tile ----
        if (more) {
            const int nb = buf ^ 1;
            *(v2f*)&As[nb][kpA + 0][mA][0] = (v2f){ra0.x, ra0.y};
            *(v2f*)&As[nb][kpA + 1][mA][0] = (v2f){ra0.z, ra0.w};
            *(v2f*)&As[nb][kpA + 2][mA][0] = (v2f){ra1.x, ra1.y};
            *(v2f*)&As[nb][kpA + 3][mA][0] = (v2f){ra1.z, ra1.w};
            float4 bq = {rb0.x, rb1.x, rb0.y, rb1.y};
            *(float4*)&Bs[nb][kpB][n2B][0] = bq;
            __syncthreads();
        }
    }

    // ---- epilogue: 16x16 C layout, VGPR r -> (M=r | r+8, N=lane&15) ----
#pragma unroll
    for (int i = 0; i < 2; ++i) {
#pragma unroll
        for (int j = 0; j < 2; ++j) {
            const int col     = nBlock + wn * 32 + j * 16 + ml;
            const int rowBase = mBlock + wm * 32 + i * 16 + hi * 8;
            const float bb = BIAS ? bias[col] : 0.0f;
#pragma unroll
            for (int r = 0; r < 8; ++r) {
                const size_t off = (size_t)(rowBase + r) * ldc + col;
                float v = acc[i][j][r] + bb;
                if (ACCUM) v += C[off];
                if (TANH)  v = tanhf(v);
                C[off] = v;
            }
        }
    }
}

// ---------------------------------------------------------------------------
// Row-normalized adjacency: one block per row, 8 cols/thread in registers
// ---------------------------------------------------------------------------
__global__ __launch_bounds__(256)
void build_adj_kernel(const int* __restrict__ spk, const float* __restrict__ mm,
                      float* __restrict__ adj)
{
    const int i   = blockIdx.x;
    const int tid = threadIdx.x;
    __shared__ float wsum[8];
    __shared__ float stot;

    const int   si  = spk[i];
    const float mi0 = mm[i * 3 + 0], mi1 = mm[i * 3 + 1], mi2 = mm[i * 3 + 2];

    float vals[8];
    float sum = 0.0f;
#pragma unroll
    for (int t = 0; t < 8; ++t) {
        const int   j    = tid + t * 256;
        const float tw   = expf(-0.1f * fabsf((float)(i - j)));
        const float msim = 1.0f - (fabsf(mi0 - mm[j * 3 + 0]) +
                                   fabsf(mi1 - mm[j * 3 + 1]) +
                                   fabsf(mi2 - mm[j * 3 + 2])) * (1.0f / 3.0f);
        float v = (si == spk[j]) ? (0.8f * tw) : (0.5f * tw * msim);
        if (j == i) v = 1.0f;
        vals[t] = v;
        sum += v;
    }
#pragma unroll
    for (int off = 16; off > 0; off >>= 1)
        sum += __shfl_down(sum, off, 32);
    if ((tid & 31) == 0) wsum[tid >> 5] = sum;
    __syncthreads();
    if (tid == 0) {
        float s = 0.0f;
        for (int w = 0; w < 8; ++w) s += wsum[w];
        stot = 1.0f / (s + 1e-8f);
    }
    __syncthreads();
    const float r = stot;
#pragma unroll
    for (int t = 0; t < 8; ++t)
        adj[(size_t)i * B_DIM + tid + t * 256] = vals[t] * r;
}

// out = X + alpha * Y   (vectorized; also used as copy with alpha=0)
__global__ __launch_bounds__(256)
void axpy4_kernel(float* __restrict__ out, const float* __restrict__ X,
                  const float* __restrict__ Y, float alpha, int n4)
{
    const int idx = blockIdx.x * blockDim.x + threadIdx.x;
    if (idx < n4) {
        float4 x = ((const float4*)X)[idx];
        float4 y = ((const float4*)Y)[idx];
        float4 o;
        o.x = x.x + alpha * y.x;
        o.y = x.y + alpha * y.y;
        o.z = x.z + alpha * y.z;
        o.w = x.w + alpha * y.w;
        ((float4*)out)[idx] = o;
    }
}

// ---------------------------------------------------------------------------
// Host side
// ---------------------------------------------------------------------------
static void rhs_eval(const float* adj, const float* src,
                     const float* W1, const float* b1,
                     const float* W2, const float* b2,
                     float* hn, float* u, float* dst, hipStream_t stream)
{
    dim3 blk(256);
    dim3 grd(H_DIM / 64, B_DIM / 128);
    // hn = adj @ src
    gemm_f32_wmma<false, false, false><<<grd, blk, 0, stream>>>(
        adj, B_DIM, src, H_DIM, nullptr, hn, H_DIM, B_DIM, H_DIM, B_DIM);
    // u = src @ W1[:H]
    gemm_f32_wmma<false, false, false><<<grd, blk, 0, stream>>>(
        src, H_DIM, W1, H_DIM, nullptr, u, H_DIM, B_DIM, H_DIM, H_DIM);
    // u = tanh(hn @ W1[H:] + u + b1)
    gemm_f32_wmma<true, true, true><<<grd, blk, 0, stream>>>(
        hn, H_DIM, W1 + (size_t)H_DIM * H_DIM, H_DIM, b1, u, H_DIM,
        B_DIM, H_DIM, H_DIM);
    // dst = u @ W2 + b2
    gemm_f32_wmma<true, false, false><<<grd, blk, 0, stream>>>(
        u, H_DIM, W2, H_DIM, b2, dst, H_DIM, B_DIM, H_DIM, H_DIM);
}

extern "C" void kernel_launch(void* const* d_in, const int* in_sizes, int n_in,
                              void* d_out, int out_size, void* d_ws, size_t ws_size,
                              hipStream_t stream)
{
    const float* features = (const float*)d_in[0];
    const int*   spk      = (const int*)  d_in[1];
    const float* mm       = (const float*)d_in[2];
    const float* W_proj   = (const float*)d_in[3];
    const float* b_proj   = (const float*)d_in[4];
    const float* W1       = (const float*)d_in[5];
    const float* b1       = (const float*)d_in[6];
    const float* W2       = (const float*)d_in[7];
    const float* b2       = (const float*)d_in[8];

    float* h  = (float*)d_out;                       // live state, 2048x1024
    float* ws = (float*)d_ws;
    const size_t BH = (size_t)B_DIM * H_DIM;
    float* adj = ws;                                 // 2048x2048
    float* y   = adj + (size_t)B_DIM * B_DIM;        // stage input
    float* hn  = y   + BH;                           // adj @ y
    float* u   = hn  + BH;                           // tanh activations
    float* kv  = u   + BH;                           // current RK slope
    float* acc = kv  + BH;                           // weighted slope sum

    dim3 blk(256);

    // adjacency
    build_adj_kernel<<<B_DIM, blk, 0, stream>>>(spk, mm, adj);

    // h0 = features @ W_proj + b_proj
    {
        dim3 grd(H_DIM / 64, B_DIM / 128);
        gemm_f32_wmma<true, false, false><<<grd, blk, 0, stream>>>(
            features, DIN, W_proj, H_DIM, b_proj, h, H_DIM, B_DIM, H_DIM, DIN);
    }

    const int   n4 = (int)(BH / 4);
    const dim3  gEW((n4 + 255) / 256);
    const float dt = 1.0f / (float)NSTEP;

    for (int s = 0; s < NSTEP; ++s) {
        // k1
        rhs_eval(adj, h, W1, b1, W2, b2, hn, u, kv, stream);
        axpy4_kernel<<<gEW, blk, 0, stream>>>(acc, kv, kv, 0.0f, n4);    // acc = k1
        axpy4_kernel<<<gEW, blk, 0, stream>>>(y, h, kv, 0.5f * dt, n4);  // y = h + dt/2 k1
        // k2
        rhs_eval(adj, y, W1, b1, W2, b2, hn, u, kv, stream);
        axpy4_kernel<<<gEW, blk, 0, stream>>>(acc, acc, kv, 2.0f, n4);   // acc += 2 k2
        axpy4_kernel<<<gEW, blk, 0, stream>>>(y, h, kv, 0.5f * dt, n4);  // y = h + dt/2 k2
        // k3
        rhs_eval(adj, y, W1, b1, W2, b2, hn, u, kv, stream);
        axpy4_kernel<<<gEW, blk, 0, stream>>>(acc, acc, kv, 2.0f, n4);   // acc += 2 k3
        axpy4_kernel<<<gEW, blk, 0, stream>>>(y, h, kv, dt, n4);         // y = h + dt k3
        // k4
        rhs_eval(adj, y, W1, b1, W2, b2, hn, u, kv, stream);
        axpy4_kernel<<<gEW, blk, 0, stream>>>(acc, acc, kv, 1.0f, n4);   // acc += k4
        axpy4_kernel<<<gEW, blk, 0, stream>>>(h, h, acc, dt / 6.0f, n4); // h += dt/6 acc
    }
}